// AffineLabelAttention_66279935312115
// MI455X (gfx1250) — compile-verified
//
#include <hip/hip_runtime.h>
#include <hip/hip_bf16.h>

typedef __attribute__((ext_vector_type(2))) float v2f;
typedef __attribute__((ext_vector_type(4))) float v4f;
typedef __attribute__((ext_vector_type(8))) float v8f;

#define BB 4
#define SS 1024
#define DD 768
#define LL 32

// ---------------------------------------------------------------------------
// Kernel 1: scores via V_WMMA_F32_16X16X4_F32.
// One wave (32 threads) per block computes a 16(i) x 32(l) tile of both
// h_score (head x W_h, + bias folded in) and d_score (dep x W_d).
// A 16x4 f32 frag: lanes 0-15 -> M=lane, K=0..1 ; lanes 16-31 -> M=lane-16, K=2..3
// B 4x16 f32 frag: lanes 0-15 -> N=lane, K=0..1 ; lanes 16-31 -> N=lane-16, K=2..3
// C/D 16x16 f32:   VGPR r -> M = r + 8*(lane>=16), N = lane & 15
// ---------------------------------------------------------------------------
__global__ __launch_bounds__(32) void score_wmma_kernel(
    const float* __restrict__ head, const float* __restrict__ dep,
    const float* __restrict__ W,    const float* __restrict__ bias,
    float* __restrict__ hsc, float* __restrict__ dsc)
{
    const int blk  = blockIdx.x;
    const int b    = blk / (SS / 16);
    const int i0   = (blk % (SS / 16)) * 16;
    const int lane = threadIdx.x;
    const int half = lane >> 4;     // 0 or 1 (selects K=0..1 vs K=2..3)
    const int l16  = lane & 15;

    const float* Ah = head + ((size_t)b * SS + (i0 + l16)) * DD + 2 * half;
    const float* Ad = dep  + ((size_t)b * SS + (i0 + l16)) * DD + 2 * half;
    const float* Wr0 = W + (size_t)(l16)      * (2 * DD) + 2 * half; // labels 0..15
    const float* Wr1 = W + (size_t)(l16 + 16) * (2 * DD) + 2 * half; // labels 16..31

    v8f ch0 = {}, ch1 = {}, cd0 = {}, cd1 = {};

    for (int k = 0; k < DD; k += 4) {
        v2f ah  = *(const v2f*)(Ah + k);
        v2f ad  = *(const v2f*)(Ad + k);
        v2f bh0 = *(const v2f*)(Wr0 + k);        // W_h rows, labels 0..15
        v2f bh1 = *(const v2f*)(Wr1 + k);        // W_h rows, labels 16..31
        v2f bd0 = *(const v2f*)(Wr0 + DD + k);   // W_d rows, labels 0..15
        v2f bd1 = *(const v2f*)(Wr1 + DD + k);   // W_d rows, labels 16..31

        ch0 = __builtin_amdgcn_wmma_f32_16x16x4_f32(false, ah, false, bh0,
                                                    (short)0, ch0, false, false);
        ch1 = __builtin_amdgcn_wmma_f32_16x16x4_f32(false, ah, false, bh1,
                                                    (short)0, ch1, false, false);
        cd0 = __builtin_amdgcn_wmma_f32_16x16x4_f32(false, ad, false, bd0,
                                                    (short)0, cd0, false, false);
        cd1 = __builtin_amdgcn_wmma_f32_16x16x4_f32(false, ad, false, bd1,
                                                    (short)0, cd1, false, false);
    }

    // Fold label bias into h_score so phase 2 is a pure add.
    const float bias0 = bias[l16];
    const float bias1 = bias[l16 + 16];

    #pragma unroll
    for (int r = 0; r < 8; ++r) {
        const int i = i0 + r + 8 * half;                 // M = r + 8*half
        hsc[((size_t)b * LL + l16)      * SS + i] = ch0[r] + bias0;
        hsc[((size_t)b * LL + l16 + 16) * SS + i] = ch1[r] + bias1;
        dsc[((size_t)b * LL + l16)      * SS + i] = cd0[r];
        dsc[((size_t)b * LL + l16 + 16) * SS + i] = cd1[r];
    }
}

// ---------------------------------------------------------------------------
// Kernel 2: out[b,l,i,j] = hsc[b,l,i] + dsc[b,l,j]   (bias already in hsc).
// 512 MiB of output -> pure store-bandwidth bound. Each block: 16 rows of one
// (b,l) tile; each thread loads its float4 of dsc ONCE and streams 16
// non-temporal B128 stores (output >> L2, keep L2 for the reused score rows).
// ---------------------------------------------------------------------------
#define ROWS 16

__global__ __launch_bounds__(256) void bcast_add_kernel(
    const float* __restrict__ hsc, const float* __restrict__ dsc,
    float* __restrict__ out)
{
    const int bl = blockIdx.x / (SS / ROWS);          // b*L + l : 0..127
    const int i0 = (blockIdx.x % (SS / ROWS)) * ROWS;
    const int t  = threadIdx.x;                       // 0..255 -> j4 position

    const v4f d4 = *(const v4f*)(dsc + (size_t)bl * SS + 4 * t);
    const float* hrow = hsc + (size_t)bl * SS + i0;
    float* orow = out + ((size_t)bl * SS + i0) * SS + 4 * t;

    #pragma unroll
    for (int r = 0; r < ROWS; ++r) {
        const float h = hrow[r];
        v4f v = d4 + h;                               // {h+d4.x, ...}
        __builtin_nontemporal_store(v, (v4f*)(orow + (size_t)r * SS));
    }
}

extern "C" void kernel_launch(void* const* d_in, const int* in_sizes, int n_in,
                              void* d_out, int out_size, void* d_ws, size_t ws_size,
                              hipStream_t stream) {
    (void)in_sizes; (void)n_in; (void)out_size; (void)ws_size;

    const float* head  = (const float*)d_in[0];   // (4,1024,768) f32
    const float* dep   = (const float*)d_in[1];   // (4,1024,768) f32
    const float* W     = (const float*)d_in[2];   // (32,1536)    f32
    const float* bias  = (const float*)d_in[3];   // (32,)        f32
    float*       out   = (float*)d_out;           // (4,32,1024,1024) f32

    float* hsc = (float*)d_ws;                    // (4,32,1024) f32 = 512 KiB
    float* dsc = hsc + (size_t)BB * LL * SS;      // (4,32,1024) f32 = 512 KiB

    // Phase 1: WMMA scores (tiny; 256 single-wave blocks).
    score_wmma_kernel<<<BB * (SS / 16), 32, 0, stream>>>(head, dep, W, bias, hsc, dsc);

    // Phase 2: 512 MiB streaming broadcast-add (store-bandwidth bound).
    bcast_add_kernel<<<(BB * LL) * (SS / ROWS), 256, 0, stream>>>(hsc, dsc, out);
}